// LogSignature_42004780155237
// MI455X (gfx1250) — compile-verified
//
#include <hip/hip_runtime.h>

typedef float v2f __attribute__((ext_vector_type(2)));
typedef float v8f __attribute__((ext_vector_type(8)));

#define B_      128
#define L_      257
#define C_      8
#define NSTEP   256
#define OUTPB   4680      // 8 + 64 + 512 + 4096

// ---- LDS layout (float offsets). Scan-phase arrays overlap the S region,
// which is only written after the scan finishes. ----
#define PATH_OFF 0        // 2056 floats, scan phase only
#define V_OFF    2060
#define H2_OFF   2068     // v/2
#define H3_OFF   2076     // v/3
#define G2_OFF   2084     // sig1 + v/2
#define G3_OFF   2092     // sig1 + v/3
#define G4_OFF   2100     // sig1 + v/4
#define T2_OFF   2112     // 64
#define U2_OFF   2176     // 64   (ends at 2240 < 4680, safe)
// post-scan regions (S = signature, P/Q = tensor-power ping-pong)
#define S1O 0
#define S2O 8
#define S3O 72
#define S4O 584
#define SREG 0
#define PREG 4680
#define QREG 9360
#define LDS_FLOATS 14040  // 56160 bytes

__global__ __launch_bounds__(32)
void logsig_scan_kernel(const float* __restrict__ path, float* __restrict__ out) {
  __shared__ float buf[LDS_FLOATS];
  const int lane = threadIdx.x;           // wave32: one wave per block
  const int b    = blockIdx.x;            // one batch element per wave
  const float* gp = path + (size_t)b * (L_ * C_);

  // ---- stage the whole path for this batch element into LDS (8.2 KB) ----
  {
    const float4* g4 = (const float4*)gp;
    float4* l4 = (float4*)&buf[PATH_OFF];
    for (int idx = lane; idx < (L_ * C_) / 4; idx += 32) l4[idx] = g4[idx];
  }
  __syncthreads();

  // ---- register-resident signature state ----
  v8f  cacc[16];            // level 4: 16 WMMA C/D tiles (256x16 view), 128 VGPRs
  v2f  a3[16];              // level 3: stored directly in WMMA A layout (lanes 16-31 dup)
  float r2[2] = {0.f, 0.f}; // level 2: lane owns ij = lane, lane+32
  float sig1  = 0.f;        // level 1: lanes 0..7 own component `lane`
#pragma unroll
  for (int T = 0; T < 16; ++T) {
    a3[T].x = 0.f; a3[T].y = 0.f;
#pragma unroll
    for (int q = 0; q < 8; ++q) cacc[T][q] = 0.f;
  }

  float pprev = 0.f;
  if (lane < C_) pprev = buf[PATH_OFF + lane];

  const int jj  = lane & 7;          // j index for this lane's sig2 elements
  const int i0  = lane >> 3;         // i for ij = lane
  const int i1  = 4 + i0;            // i for ij = lane + 32
  const int kk2 = 2 * (lane & 3);    // k pattern for WMMA-A layout (K0 even, K1 odd)
  const int ta  = (lane >> 2) & 3;   // which of 4 (ij) rows inside a tile

  // ================= Chen scan: sig <- sig (x) exp(v), 256 steps =================
  for (int t = 1; t <= NSTEP; ++t) {
    if (lane < C_) {
      float pc = buf[PATH_OFF + t * C_ + lane];
      float vv = pc - pprev; pprev = pc;
      buf[V_OFF  + lane] = vv;
      buf[H2_OFF + lane] = 0.5f * vv;
      buf[H3_OFF + lane] = vv * (1.0f / 3.0f);
      buf[G2_OFF + lane] = sig1 + 0.5f * vv;
      buf[G3_OFF + lane] = sig1 + vv * (1.0f / 3.0f);
      buf[G4_OFF + lane] = sig1 + 0.25f * vv;
      sig1 += vv;                     // level-1 update
    }
    __syncthreads();

    // per-step lane-indexed broadcasts
    float vj  = buf[V_OFF  + jj];
    float h2j = buf[H2_OFF + jj];
    float h3j = buf[H3_OFF + jj];
    float g2a = buf[G2_OFF + i0], g2b = buf[G2_OFF + i1];
    float g3a = buf[G3_OFF + i0], g3b = buf[G3_OFF + i1];
    float g4a = buf[G4_OFF + i0], g4b = buf[G4_OFF + i1];
    float hkA = buf[H2_OFF + kk2], hkB = buf[H2_OFF + kk2 + 1];
    float vkA = buf[V_OFF  + kk2], vkB = buf[V_OFF  + kk2 + 1];

    // Horner temporaries at level 2 (from OLD sig), then level-2 update
    float t2a = r2[0] + h3j * g4a;    // feeds level-4 chain
    float t2b = r2[1] + h3j * g4b;
    float u2a = r2[0] + h2j * g3a;    // feeds level-3 update
    float u2b = r2[1] + h2j * g3b;
    r2[0] += vj * g2a;
    r2[1] += vj * g2b;
    buf[T2_OFF + lane]      = t2a;
    buf[T2_OFF + lane + 32] = t2b;
    buf[U2_OFF + lane]      = u2a;
    buf[U2_OFF + lane + 32] = u2b;
    __syncthreads();

    // B operand: rank-2 (K rows 0,1 = v split over the 16 columns; rows 2,3 = 0)
    v2f Bv;
    Bv.x = (lane < 8) ? vj : 0.0f;
    Bv.y = (lane >= 8 && lane < 16) ? vj : 0.0f;

    // level-3 Horner + WMMA level-4 rank-2 accumulate + level-3 update
#pragma unroll
    for (int T = 0; T < 16; ++T) {
      float t2v = buf[T2_OFF + 4 * T + ta];
      float u2v = buf[U2_OFF + 4 * T + ta];
      v2f Av;                           // t3 in A layout (K=0 even k, K=1 odd k)
      Av.x = a3[T].x + hkA * t2v;
      Av.y = a3[T].y + hkB * t2v;
      cacc[T] = __builtin_amdgcn_wmma_f32_16x16x4_f32(
          false, Av, false, Bv, (short)0, cacc[T], false, false);
      a3[T].x += vkA * u2v;             // sig3 += v_k * u2
      a3[T].y += vkB * u2v;
    }
    __syncthreads();
  }

  // ================= dump signature S (and P = S) to LDS =================
  if (lane < 8) { buf[SREG + S1O + lane] = sig1; buf[PREG + S1O + lane] = sig1; }
  buf[SREG + S2O + lane]      = r2[0];  buf[PREG + S2O + lane]      = r2[0];
  buf[SREG + S2O + 32 + lane] = r2[1];  buf[PREG + S2O + 32 + lane] = r2[1];
  if (lane < 16) {
#pragma unroll
    for (int T = 0; T < 16; ++T) {
      int f = 32 * T + 2 * lane;
      buf[SREG + S3O + f]     = a3[T].x;  buf[PREG + S3O + f]     = a3[T].x;
      buf[SREG + S3O + f + 1] = a3[T].y;  buf[PREG + S3O + f + 1] = a3[T].y;
    }
  }
  {
    const int hi = (lane >> 4) * 8, cc = lane & 15;
#pragma unroll
    for (int T = 0; T < 16; ++T)
#pragma unroll
      for (int q = 0; q < 8; ++q) {
        int f = (16 * T + q + hi) * 16 + cc;
        buf[SREG + S4O + f] = cacc[T][q];
        buf[PREG + S4O + f] = cacc[T][q];
      }
  }
  __syncthreads();

  // ================= log(1+S): result += (-1)^{n+1}/n * S^{(x)n} =================
  int Pb = PREG, Qb = QREG;
  const int mm = lane & 15;
  for (int n = 2; n <= 4; ++n) {
    float coef = ((n & 1) ? 1.0f : -1.0f) / (float)n;
    if (lane < 8) buf[Qb + S1O + lane] = 0.0f;          // level-1 power is 0 for n>=2
    // level 2: P1 (x) S1
#pragma unroll
    for (int h = 0; h < 2; ++h) {
      int ij = lane + 32 * h;
      float q = buf[Pb + S1O + (ij >> 3)] * buf[SREG + S1O + (ij & 7)];
      r2[h] += coef * q;
      buf[Qb + S2O + ij] = q;
    }
    // level 3: P1 (x) S2 + P2 (x) S1   (same element mapping as a3 registers)
#pragma unroll
    for (int T = 0; T < 16; ++T) {
#pragma unroll
      for (int r = 0; r < 2; ++r) {
        int f = 32 * T + 2 * mm + r;
        float q = buf[Pb + S1O + (f >> 6)] * buf[SREG + S2O + (f & 63)]
                + buf[Pb + S2O + (f >> 3)] * buf[SREG + S1O + (f & 7)];
        if (r == 0) a3[T].x += coef * q; else a3[T].y += coef * q;
        if (lane < 16) buf[Qb + S3O + f] = q;
      }
    }
    // level 4: P1 (x) S3 + P2 (x) S2 + P3 (x) S1   (WMMA C-tile element mapping)
    {
      const int hi = (lane >> 4) * 8, cc = lane & 15;
#pragma unroll
      for (int T = 0; T < 16; ++T)
#pragma unroll
        for (int q8 = 0; q8 < 8; ++q8) {
          int f = (16 * T + q8 + hi) * 16 + cc;
          float q = buf[Pb + S1O + (f >> 9)] * buf[SREG + S3O + (f & 511)]
                  + buf[Pb + S2O + (f >> 6)] * buf[SREG + S2O + (f & 63)]
                  + buf[Pb + S3O + (f >> 3)] * buf[SREG + S1O + (f & 7)];
          cacc[T][q8] += coef * q;
          buf[Qb + S4O + f] = q;
        }
    }
    __syncthreads();
    int tmp = Pb; Pb = Qb; Qb = tmp;    // P <- newP
  }

  // ================= write output: concat(level1..level4) per batch =================
  float* ob = out + (size_t)b * OUTPB;
  if (lane < 8) ob[lane] = sig1;
  ob[8 + lane]      = r2[0];
  ob[8 + 32 + lane] = r2[1];
  if (lane < 16) {
#pragma unroll
    for (int T = 0; T < 16; ++T) {
      int f = 32 * T + 2 * lane;
      ob[72 + f]     = a3[T].x;
      ob[72 + f + 1] = a3[T].y;
    }
  }
  {
    const int hi = (lane >> 4) * 8, cc = lane & 15;
#pragma unroll
    for (int T = 0; T < 16; ++T)
#pragma unroll
      for (int q = 0; q < 8; ++q) {
        int f = (16 * T + q + hi) * 16 + cc;
        ob[584 + f] = cacc[T][q];
      }
  }
}

extern "C" void kernel_launch(void* const* d_in, const int* in_sizes, int n_in,
                              void* d_out, int out_size, void* d_ws, size_t ws_size,
                              hipStream_t stream) {
  const float* path = (const float*)d_in[0];
  float* out = (float*)d_out;
  logsig_scan_kernel<<<dim3(B_), dim3(32), 0, stream>>>(path, out);
}